// ComplementaryLIFNeuron_57226144252674
// MI455X (gfx1250) — compile-verified
//
#include <hip/hip_runtime.h>

// Complementary LIF neuron forward pass (spikes only), CDNA5 / gfx1250.
//
// x_seq is [(step*b), n, c] contiguous; viewing it as [step, N] with
// N = b*n*c makes the timestep stride exactly N floats, and the output uses
// the identical layout. The op is a per-neuron sequential recurrence over t
// (Heaviside output feeds the next state) -> elementwise, zero matrix work,
// purely bandwidth bound: ~77 MB read + ~77 MB write @ 23.3 TB/s ~ 6.6 us.
//
// Strategy: one thread per 4 channels (b128), all `step` loads hoisted ahead
// of the serial recurrence (one s_wait_loadcnt), non-temporal TH on both
// loads and stores (every byte touched exactly once), wave32 256-thread
// blocks, and pure 32-bit index math (in_sizes is int, so total < 2^31).

typedef float v4f __attribute__((ext_vector_type(4)));

#define TAU_INV   0.5f   // 1/TAU
#define DECAY     0.5f   // 1 - 1/TAU
#define V_TH      1.0f
#define LOG2E     1.4426950408889634f

__device__ __forceinline__ float fast_sigmoid(float x) {
    // 1 / (1 + exp(-x)) via v_exp_f32 (base-2) + v_rcp_f32
    float e = __builtin_amdgcn_exp2f(x * -LOG2E);
    return __builtin_amdgcn_rcpf(1.0f + e);
}

// One LIF timestep on a 4-wide register vector. Updates v, m; returns spikes.
__device__ __forceinline__ v4f lif_step4(v4f x, v4f& v, v4f& m) {
    v4f s;
#pragma unroll
    for (int k = 0; k < 4; ++k) {
        float vv = v[k] * DECAY + x[k];               // neuronal charge
        float mm = m[k] * fast_sigmoid(vv * TAU_INV); // complementary membrane
        float sk = (vv >= V_TH) ? 1.0f : 0.0f;        // Heaviside fire
        mm += sk;
        vv -= sk * V_TH;                              // soft reset
        vv -= sk * fast_sigmoid(mm);                  // complementary reset
        v[k] = vv;
        m[k] = mm;
        s[k] = sk;
    }
    return s;
}

__global__ __launch_bounds__(256) void clif_fwd_kernel(
    const float* __restrict__ x, float* __restrict__ out,
    const int* __restrict__ step_p, unsigned int total)
{
    const unsigned int step    = (unsigned int)*step_p;  // uniform scalar load
    const unsigned int n_elems = total / step;           // 32-bit udiv (short)
    const unsigned int n4      = n_elems >> 2;

    const unsigned int tid = blockIdx.x * blockDim.x + threadIdx.x;

    const v4f* __restrict__ x4 = (const v4f*)x;
    v4f* __restrict__ o4 = (v4f*)out;

    if (tid < n4) {
        const unsigned int i = tid;
        if (step == 4u) {
            // All 4 loads issue back-to-back (global_load_b128 th:NT); the
            // serial t-chain then pays memory latency exactly once.
            v4f x0 = __builtin_nontemporal_load(&x4[i]);
            v4f x1 = __builtin_nontemporal_load(&x4[i + n4]);
            v4f x2 = __builtin_nontemporal_load(&x4[i + 2u * n4]);
            v4f x3 = __builtin_nontemporal_load(&x4[i + 3u * n4]);

            v4f v = {0.f, 0.f, 0.f, 0.f};
            v4f m = {0.f, 0.f, 0.f, 0.f};
            v4f s0 = lif_step4(x0, v, m);
            v4f s1 = lif_step4(x1, v, m);
            v4f s2 = lif_step4(x2, v, m);
            v4f s3 = lif_step4(x3, v, m);

            __builtin_nontemporal_store(s0, &o4[i]);
            __builtin_nontemporal_store(s1, &o4[i + n4]);
            __builtin_nontemporal_store(s2, &o4[i + 2u * n4]);
            __builtin_nontemporal_store(s3, &o4[i + 3u * n4]);
        } else {
            // Generic step count.
            v4f v = {0.f, 0.f, 0.f, 0.f};
            v4f m = {0.f, 0.f, 0.f, 0.f};
            for (unsigned int t = 0; t < step; ++t) {
                v4f xt = __builtin_nontemporal_load(&x4[i + t * n4]);
                v4f st = lif_step4(xt, v, m);
                __builtin_nontemporal_store(st, &o4[i + t * n4]);
            }
        }
    }

    // Scalar tail (n_elems not a multiple of 4). Empty for the reference
    // shape (c = 768).
    const unsigned int rem_base = n4 << 2;
    const unsigned int rem = n_elems - rem_base;
    if (tid < rem) {
        const unsigned int j = rem_base + tid;
        float v = 0.f, m = 0.f;
        for (unsigned int t = 0; t < step; ++t) {
            float xv = x[t * n_elems + j];
            v = v * DECAY + xv;
            m *= fast_sigmoid(v * TAU_INV);
            float s = (v >= V_TH) ? 1.0f : 0.0f;
            m += s;
            v -= s * V_TH;
            v -= s * fast_sigmoid(m);
            out[t * n_elems + j] = s;
        }
    }
}

extern "C" void kernel_launch(void* const* d_in, const int* in_sizes, int n_in,
                              void* d_out, int out_size, void* d_ws, size_t ws_size,
                              hipStream_t stream) {
    const float* x      = (const float*)d_in[0];
    const int*   step_p = (const int*)d_in[1];
    float*       out    = (float*)d_out;

    const unsigned int total = (unsigned int)in_sizes[0];

    // `step` lives on-device; size the grid as an upper bound (step >= 1):
    // at most total/4 vec4 work items per timestep slice. Excess threads see
    // tid >= n4 and exit immediately (cheap, deterministic).
    const unsigned int threads = 256;
    unsigned int max_vec = (total + 3u) / 4u;
    unsigned int blocks  = (max_vec + threads - 1u) / threads;
    if (blocks < 1u) blocks = 1u;

    clif_fwd_kernel<<<dim3(blocks), dim3(threads), 0, stream>>>(
        x, out, step_p, total);
}